// CRF_32117765439826
// MI455X (gfx1250) — compile-verified
//
#include <hip/hip_runtime.h>
#include <hip/hip_bf16.h>
#include <cstdint>

#define NT 48
#define BB 1024
#define SS 512

typedef __attribute__((ext_vector_type(2))) float v2f;
typedef __attribute__((ext_vector_type(8))) float v8f;

// ---------------- kernel 1: per-batch gold path score ----------------
// score[b] = emit[0] + sum_{s>=1} emit[s]*mask[s] + sum_{s>=1} T[tag[s-1],tag[s]]*mask[s]
__global__ __launch_bounds__(128) void crf_score_kernel(
    const float* __restrict__ em, const float* __restrict__ trans,
    const int* __restrict__ tags, const int* __restrict__ mask,
    float* __restrict__ score) {
  const int b = blockIdx.x;
  const int tid = threadIdx.x;
  const float* emb = em + (size_t)b * SS * NT;
  const int* tgb = tags + (size_t)b * SS;
  const int* mkb = mask + (size_t)b * SS;
  float acc = 0.0f;
  for (int s = tid; s < SS; s += 128) {
    const int t = tgb[s];
    const float m = (s == 0) ? 1.0f : (float)mkb[s];
    acc += emb[s * NT + t] * m;
    if (s >= 1) acc += trans[tgb[s - 1] * NT + t] * (float)mkb[s];
  }
  __shared__ float red[128];
  red[tid] = acc;
  __syncthreads();
  for (int off = 64; off > 0; off >>= 1) {
    if (tid < off) red[tid] += red[tid + off];
    __syncthreads();
  }
  if (tid == 0) score[b] = red[0];
}

// ---------------- kernel 2: forward algorithm via WMMA ----------------
// One wave owns 16 batch rows. Alphas live in WMMA C-layout registers
// (3 tiles of v8f covering 48 tags). Each scan step:
//   m = rowmax(alpha); P = exp(alpha - m)         (C-layout, TRANS ops)
//   P -> LDS -> A-fragments (transpose)           (intra-wave DS, in-order)
//   D = P @ expT  : 12 K-chunks x 3 N-tiles of v_wmma_f32_16x16x4_f32
//   alpha = mask ? log(D) + m + emission : alpha
__global__ __launch_bounds__(256) void crf_forward_kernel(
    const float* __restrict__ em, const float* __restrict__ trans,
    const int* __restrict__ mask, float* __restrict__ lp) {
  __shared__ float sExpT[NT * NT];      // exp(T), [K][N] row-major
  __shared__ float sP[8][16 * NT];      // per-wave transpose staging

  const int tid = threadIdx.x;
  for (int i = tid; i < NT * NT; i += 256) sExpT[i] = __expf(trans[i]);
  __syncthreads();

  const int wid = tid >> 5;
  const int lane = tid & 31;
  const int half = lane >> 4;   // 0: lanes 0-15, 1: lanes 16-31
  const int ln = lane & 15;
  const int rowBase = (blockIdx.x * 8 + wid) * 16;
  float* __restrict__ myP = &sP[wid][0];

  // Hoist all B fragments (expT) into VGPRs: [kchunk][ntile][kreg].
  // B 4x16 layout assumed to mirror A's K striping: lane holds N=ln,
  // K = 4*k + 2*half + reg.
  float bf[12][3][2];
#pragma unroll
  for (int k = 0; k < 12; ++k) {
    const int kk = k * 4 + half * 2;
#pragma unroll
    for (int t = 0; t < 3; ++t) {
      bf[k][t][0] = sExpT[(kk + 0) * NT + t * 16 + ln];
      bf[k][t][1] = sExpT[(kk + 1) * NT + t * 16 + ln];
    }
  }

  // alpha init: tag 0 -> 0, others -> -10000. C-layout: element r of tile t
  // corresponds to (row = rowBase + r + 8*half, tag = 16*t + ln).
  v8f a0, a1, a2;
#pragma unroll
  for (int r = 0; r < 8; ++r) {
    a0[r] = (ln == 0) ? 0.0f : -10000.0f;
    a1[r] = -10000.0f;
    a2[r] = -10000.0f;
  }

  for (int s = 0; s < SS; ++s) {
    // ---- row max over 48 tags (xor 1/2/4/8 reduces within 16-lane halves)
    float rmax[8];
#pragma unroll
    for (int r = 0; r < 8; ++r) {
      float v = fmaxf(fmaxf(a0[r], a1[r]), a2[r]);
      v = fmaxf(v, __shfl_xor(v, 1, 32));
      v = fmaxf(v, __shfl_xor(v, 2, 32));
      v = fmaxf(v, __shfl_xor(v, 4, 32));
      v = fmaxf(v, __shfl_xor(v, 8, 32));
      rmax[r] = v;
    }
    // ---- P = exp(alpha - rowmax) -> LDS [row][tag] (transpose staging)
#pragma unroll
    for (int r = 0; r < 8; ++r) {
      const int M = r + half * 8;
      myP[M * NT + 0  + ln] = __expf(a0[r] - rmax[r]);
      myP[M * NT + 16 + ln] = __expf(a1[r] - rmax[r]);
      myP[M * NT + 32 + ln] = __expf(a2[r] - rmax[r]);
    }
    // (same-wave DS ops are in-order; no barrier needed)

    // ---- D = P @ expT via 36 WMMAs
    v8f d0 = {}, d1 = {}, d2 = {};
#pragma unroll
    for (int k = 0; k < 12; ++k) {
      const int kk = k * 4 + half * 2;   // A 16x4 f32: lanes 0-15 hold K0/K1, 16-31 hold K2/K3
      v2f a;
      a.x = myP[ln * NT + kk + 0];
      a.y = myP[ln * NT + kk + 1];
      v2f b0; b0.x = bf[k][0][0]; b0.y = bf[k][0][1];
      v2f b1; b1.x = bf[k][1][0]; b1.y = bf[k][1][1];
      v2f b2; b2.x = bf[k][2][0]; b2.y = bf[k][2][1];
      d0 = __builtin_amdgcn_wmma_f32_16x16x4_f32(false, a, false, b0, (short)0, d0, false, false);
      d1 = __builtin_amdgcn_wmma_f32_16x16x4_f32(false, a, false, b1, (short)0, d1, false, false);
      d2 = __builtin_amdgcn_wmma_f32_16x16x4_f32(false, a, false, b2, (short)0, d2, false, false);
    }

    // ---- alpha_new = log(D) + rowmax + emission ; masked update
#pragma unroll
    for (int r = 0; r < 8; ++r) {
      const int row = rowBase + r + half * 8;
      const float* ep = em + ((size_t)row * SS + s) * NT;
      const float mv = (float)mask[(size_t)row * SS + s];
      const float omv = 1.0f - mv;
      const float n0 = __logf(d0[r]) + rmax[r] + ep[0 + ln];
      const float n1 = __logf(d1[r]) + rmax[r] + ep[16 + ln];
      const float n2 = __logf(d2[r]) + rmax[r] + ep[32 + ln];
      a0[r] = mv * n0 + omv * a0[r];
      a1[r] = mv * n1 + omv * a1[r];
      a2[r] = mv * n2 + omv * a2[r];
      if (s + 1 < SS) __builtin_prefetch(ep + NT, 0, 0);  // global_prefetch_b8
    }
  }

  // ---- log_partition[row] = logsumexp(alpha) over 48 tags
#pragma unroll
  for (int r = 0; r < 8; ++r) {
    float v = fmaxf(fmaxf(a0[r], a1[r]), a2[r]);
    v = fmaxf(v, __shfl_xor(v, 1, 32));
    v = fmaxf(v, __shfl_xor(v, 2, 32));
    v = fmaxf(v, __shfl_xor(v, 4, 32));
    v = fmaxf(v, __shfl_xor(v, 8, 32));
    float ssum = __expf(a0[r] - v) + __expf(a1[r] - v) + __expf(a2[r] - v);
    ssum += __shfl_xor(ssum, 1, 32);
    ssum += __shfl_xor(ssum, 2, 32);
    ssum += __shfl_xor(ssum, 4, 32);
    ssum += __shfl_xor(ssum, 8, 32);
    if (ln == 0) lp[rowBase + r + half * 8] = v + __logf(ssum);
  }
}

// ---------------- kernel 3: mean(lp - score) ----------------
__global__ __launch_bounds__(256) void crf_final_kernel(
    const float* __restrict__ lp, const float* __restrict__ score,
    float* __restrict__ out) {
  __shared__ float red[256];
  const int tid = threadIdx.x;
  float a = 0.0f;
  for (int b = tid; b < BB; b += 256) a += lp[b] - score[b];
  red[tid] = a;
  __syncthreads();
  for (int off = 128; off > 0; off >>= 1) {
    if (tid < off) red[tid] += red[tid + off];
    __syncthreads();
  }
  if (tid == 0) out[0] = red[0] * (1.0f / (float)BB);
}

extern "C" void kernel_launch(void* const* d_in, const int* in_sizes, int n_in,
                              void* d_out, int out_size, void* d_ws, size_t ws_size,
                              hipStream_t stream) {
  (void)in_sizes; (void)n_in; (void)out_size; (void)ws_size;
  const float* em    = (const float*)d_in[0];  // [B,S,48] f32
  const float* trans = (const float*)d_in[1];  // [48,48]  f32
  const int*   tags  = (const int*)d_in[2];    // [B,S]    i32
  const int*   mask  = (const int*)d_in[3];    // [B,S]    i32

  float* lp    = (float*)d_ws;      // [B]
  float* score = lp + BB;           // [B]

  crf_score_kernel<<<BB, 128, 0, stream>>>(em, trans, tags, mask, score);
  crf_forward_kernel<<<BB / 128, 256, 0, stream>>>(em, trans, mask, lp);
  crf_final_kernel<<<1, 256, 0, stream>>>(lp, score, (float*)d_out);
}